// HashGrid_88278757802387
// MI455X (gfx1250) — compile-verified
//
#include <hip/hip_runtime.h>
#include <stdint.h>

#ifndef __has_builtin
#define __has_builtin(x) 0
#endif

#if __has_builtin(__builtin_amdgcn_tensor_load_to_lds) && __has_builtin(__builtin_amdgcn_s_wait_tensorcnt)
#define HG_USE_TDM 1
#else
#define HG_USE_TDM 0
#endif

#define HG_CB_SIZE (1u << 19)   // CODEBOOK_SIZE = 2^19 entries per LOD
#define HG_SZ0 4913u            // 17^3
#define HG_SZ1 19683u           // 27^3

typedef unsigned int hg_v4u __attribute__((ext_vector_type(4)));
typedef int          hg_v8i __attribute__((ext_vector_type(8)));
typedef int          hg_v4i __attribute__((ext_vector_type(4)));

struct hg_f3 { float x, y, z; };   // 12B -> single global_load_b96 per thread

#if HG_USE_TDM
// Issue one TDM load of `nelem` contiguous f32 elements from global -> LDS.
// Descriptor layout per CDNA5 ISA 8.3/8.4 (D# group0 128b, group1 256b), 2D tensor:
//   tensor_dim0 = tile_dim0 = nelem, tensor_dim1 = tile_dim1 = 1, data_size = 4B.
// This toolchain's builtin takes 6 args: (v4u g0, v8i g1, v4i g2, v4i g3, v8i extra, i32 cpol).
__device__ __forceinline__ void hg_tdm_load_row(const float* __restrict__ gsrc,
                                                uint32_t lds_byte_off,
                                                uint32_t nelem /* <= 65535 */) {
  uint64_t ga = (uint64_t)(uintptr_t)gsrc;
  hg_v4u g0;
  g0.x = 1u;                                   // count=1, user descriptor, no gather mode
  g0.y = lds_byte_off;                         // lds_addr (bytes), bits 63:32
  g0.z = (uint32_t)ga;                         // global_addr[31:0], bits 95:64
  g0.w = (uint32_t)(ga >> 32) | (2u << 30);    // global_addr[56:32] | type=2 ("image")
  hg_v8i g1;
  g1[0] = (int)(2u << 16);                     // workgroup_mask=0, data_size=2 (4 bytes)
  g1[1] = (int)((nelem & 0xFFFFu) << 16);      // tensor_dim0[15:0] at bits 79:48
  g1[2] = (int)((nelem >> 16) | (1u << 16));   // tensor_dim0[31:16] | tensor_dim1=1
  g1[3] = (int)(nelem << 16);                  // tensor_dim1[31:16]=0 | tile_dim0=nelem
  g1[4] = 1;                                   // tile_dim1=1, tile_dim2=0 (2D)
  g1[5] = (int)nelem;                          // tensor_dim0_stride[31:0]
  g1[6] = 0;                                   // stride0[47:32]=0, stride1 lo=0
  g1[7] = 0;
  hg_v4i gz4 = {0, 0, 0, 0};                   // groups 2/3 unused for 2D tensor
  hg_v8i gz8 = {0, 0, 0, 0, 0, 0, 0, 0};       // extra group (zero-filled)
  __builtin_amdgcn_tensor_load_to_lds(g0, g1, gz4, gz4, gz8, 0);
}
#endif

// One LOD of hashed trilinear interpolation. RES/SIZE are compile-time so the
// modulo folds to AND (pow2 LODs) or a magic-multiply sequence.
template <int RES, unsigned SIZE>
__device__ __forceinline__ float hg_lod(float px, float py, float pz,
                                        const float* __restrict__ tab) {
  // x = (p*0.5 + 0.5) * (RES-1) = p*s + s with s = 0.5*(RES-1)
  const float s = 0.5f * (float)(RES - 1);
  float fx = __builtin_fmaf(px, s, s);
  float fy = __builtin_fmaf(py, s, s);
  float fz = __builtin_fmaf(pz, s, s);
  float cx = fminf(fmaxf(floorf(fx), 0.0f), (float)(RES - 2));
  float cy = fminf(fmaxf(floorf(fy), 0.0f), (float)(RES - 2));
  float cz = fminf(fmaxf(floorf(fz), 0.0f), (float)(RES - 2));
  float tx = fx - cx, ty = fy - cy, tz = fz - cz;
  float w0x = 1.0f - tx, w1x = tx;
  float w0y = 1.0f - ty, w1y = ty;
  float w0z = 1.0f - tz, w1z = tz;
  // hash = x*1 ^ y*2654435761 ^ z*805459861 (uint32 wraparound)
  uint32_t hx0 = (uint32_t)(int)cx;
  uint32_t hx1 = hx0 + 1u;
  uint32_t hy0 = (uint32_t)(int)cy * 2654435761u;
  uint32_t hy1 = hy0 + 2654435761u;
  uint32_t hz0 = (uint32_t)(int)cz * 805459861u;
  uint32_t hz1 = hz0 + 805459861u;

  float acc = 0.0f;
#define HG_CORNER(ox, oy, oz)                                                   \
  do {                                                                          \
    uint32_t h = ((ox) ? hx1 : hx0) ^ ((oy) ? hy1 : hy0) ^ ((oz) ? hz1 : hz0);  \
    uint32_t idx = ((SIZE & (SIZE - 1u)) == 0u) ? (h & (SIZE - 1u)) : (h % SIZE);\
    float w = ((ox) ? w1x : w0x) * ((oy) ? w1y : w0y) * ((oz) ? w1z : w0z);     \
    acc = __builtin_fmaf(w, tab[idx], acc);                                     \
  } while (0)
  // corner order matches reference _OFFS: i = (ox<<2)|(oy<<1)|oz
  HG_CORNER(0, 0, 0); HG_CORNER(0, 0, 1); HG_CORNER(0, 1, 0); HG_CORNER(0, 1, 1);
  HG_CORNER(1, 0, 0); HG_CORNER(1, 0, 1); HG_CORNER(1, 1, 0); HG_CORNER(1, 1, 1);
#undef HG_CORNER
  return acc;
}

__global__ void __launch_bounds__(256)
HashGrid_88278757802387_kernel(const float* __restrict__ pts,
                               const float* __restrict__ codebook,
                               float* __restrict__ out, int n) {
  extern __shared__ float smem[];           // [HG_SZ0 + HG_SZ1] floats (98.4 KB)
  float* t0 = smem;                         // LOD0 table (17^3)
  float* t1 = smem + HG_SZ0;                // LOD1 table (27^3)

#if HG_USE_TDM
  if (threadIdx.x < 32u) {                  // wave 0 issues both TDM DMAs
    uint32_t base = (uint32_t)(uintptr_t)smem;   // LDS byte offset of allocation
    hg_tdm_load_row(codebook + 0u * HG_CB_SIZE, base, HG_SZ0);
    hg_tdm_load_row(codebook + 1u * HG_CB_SIZE, base + HG_SZ0 * 4u, HG_SZ1);
    __builtin_amdgcn_s_wait_tensorcnt(0);   // TENSORcnt == 0 -> LDS populated
  }
#else
  for (uint32_t k = threadIdx.x; k < HG_SZ0; k += blockDim.x) t0[k] = codebook[k];
  for (uint32_t k = threadIdx.x; k < HG_SZ1; k += blockDim.x) t1[k] = codebook[HG_CB_SIZE + k];
#endif
  __syncthreads();

  int i = (int)(blockIdx.x * 256u + threadIdx.x);
  if (i >= n) return;

  const hg_f3 p = ((const hg_f3*)pts)[i];   // one b96 load instead of 3x b32

  float acc = 0.0f;
  // LODs 0-1 from LDS-staged tables
  acc += hg_lod<17, 4913u>(p.x, p.y, p.z, t0);
  acc += hg_lod<27, 19683u>(p.x, p.y, p.z, t1);
  // LODs 2-7 from global (L2-resident, 192MB L2 holds the whole codebook)
  acc += hg_lod<44, 85184u>(p.x, p.y, p.z, codebook + 2u * HG_CB_SIZE);
  acc += hg_lod<71, 357911u>(p.x, p.y, p.z, codebook + 3u * HG_CB_SIZE);
  acc += hg_lod<116, 524288u>(p.x, p.y, p.z, codebook + 4u * HG_CB_SIZE);
  acc += hg_lod<191, 524288u>(p.x, p.y, p.z, codebook + 5u * HG_CB_SIZE);
  acc += hg_lod<313, 524288u>(p.x, p.y, p.z, codebook + 6u * HG_CB_SIZE);
  acc += hg_lod<513, 524288u>(p.x, p.y, p.z, codebook + 7u * HG_CB_SIZE);

  out[i] = acc;
}

extern "C" void kernel_launch(void* const* d_in, const int* in_sizes, int n_in,
                              void* d_out, int out_size, void* d_ws, size_t ws_size,
                              hipStream_t stream) {
  const float* pts      = (const float*)d_in[0];   // [N,3] f32
  const float* codebook = (const float*)d_in[1];   // [8, 2^19, 1] f32
  float* out            = (float*)d_out;           // [N,1] f32
  (void)d_ws; (void)ws_size; (void)n_in; (void)out_size;

  const int n = in_sizes[0] / 3;
  dim3 block(256);
  dim3 grid((unsigned)((n + 255) / 256));
  size_t shmem = (size_t)(HG_SZ0 + HG_SZ1) * sizeof(float);  // 98,384 bytes
  HashGrid_88278757802387_kernel<<<grid, block, shmem, stream>>>(pts, codebook, out, n);
}